// BigramLanguageAttentionModel_89919435309163
// MI455X (gfx1250) — compile-verified
//
#include <hip/hip_runtime.h>
#include <math.h>

#define BATCH      8
#define CONTEXT    16
#define EMBED      32
#define NUM_HEADS  4
#define HEAD_SIZE  8
#define VOCAB      50257
#define ROWS       (BATCH * CONTEXT)            // 128
#define N_TILES    ((VOCAB + 15) / 16)          // 3142

typedef __attribute__((ext_vector_type(2))) float v2f;
typedef __attribute__((ext_vector_type(8))) float v8f;

// ---------------------------------------------------------------------------
// Kernel 1: x = tok_emb[idx] + pos_emb, then 4-head causal attention.
// Output X[b*16+t][h*8+d] (concat heads) into workspace. Also zeros loss slot.
// Work is trivial (< 1 MFLOP total): plain VALU, LDS staging.
// ---------------------------------------------------------------------------
__global__ __launch_bounds__(128)
void embed_attn_kernel(const int* __restrict__ idx,
                       const float* __restrict__ tok_emb,
                       const float* __restrict__ pos_emb,
                       const float* __restrict__ Wq,
                       const float* __restrict__ Wk,
                       const float* __restrict__ Wv,
                       float* __restrict__ X,
                       float* __restrict__ loss_slot) {
    __shared__ float x[CONTEXT][EMBED];
    __shared__ float q[CONTEXT][HEAD_SIZE];
    __shared__ float k[CONTEXT][HEAD_SIZE];
    __shared__ float v[CONTEXT][HEAD_SIZE];

    const int b   = blockIdx.x;
    const int tid = threadIdx.x;                 // 0..127

    if (b == 0 && tid == 0) loss_slot[0] = 0.0f; // deterministic re-init each call

    for (int i = tid; i < CONTEXT * EMBED; i += 128) {
        const int t = i / EMBED, c = i % EMBED;
        x[t][c] = tok_emb[idx[b * CONTEXT + t] * EMBED + c] + pos_emb[t * EMBED + c];
    }
    __syncthreads();

    const int t = tid / HEAD_SIZE;               // 0..15
    const int d = tid % HEAD_SIZE;               // 0..7
    const float scale = rsqrtf((float)EMBED);    // NOTE: C**-0.5 per reference

    for (int h = 0; h < NUM_HEADS; ++h) {
        const float* wq = Wq + h * EMBED * HEAD_SIZE;
        const float* wk = Wk + h * EMBED * HEAD_SIZE;
        const float* wv = Wv + h * EMBED * HEAD_SIZE;
        float aq = 0.f, ak = 0.f, av = 0.f;
        #pragma unroll
        for (int c = 0; c < EMBED; ++c) {
            const float xv = x[t][c];
            aq += xv * wq[c * HEAD_SIZE + d];
            ak += xv * wk[c * HEAD_SIZE + d];
            av += xv * wv[c * HEAD_SIZE + d];
        }
        q[t][d] = aq; k[t][d] = ak; v[t][d] = av;
        __syncthreads();

        // Online-softmax over causal positions s <= t
        float m = -INFINITY, s = 0.f, acc = 0.f;
        for (int j = 0; j <= t; ++j) {
            float sc = 0.f;
            #pragma unroll
            for (int c = 0; c < HEAD_SIZE; ++c) sc += q[t][c] * k[j][c];
            sc *= scale;
            const float mn   = fmaxf(m, sc);
            const float corr = __expf(m - mn);   // first iter: exp(-inf)=0
            const float e    = __expf(sc - mn);
            s   = s * corr + e;
            acc = acc * corr + e * v[j][d];
            m = mn;
        }
        X[(b * CONTEXT + t) * EMBED + h * HEAD_SIZE + d] = acc / s;
        __syncthreads();                         // q/k/v reused next head
    }
}

// ---------------------------------------------------------------------------
// Kernel 2: logits[128,50257] = X[128,32] @ W_lm[32,50257] + b_lm
// via V_WMMA_F32_16X16X4_F32, 8 chained WMMAs (K=32) per 16x16 tile.
// One wave per output tile; 8 waves per block; grid (8 m-tiles, 393).
// Column addresses are clamped (not predicated) so EXEC stays all-1s around
// the WMMAs; a clamped B column only pollutes its own (unstored) D column.
// ---------------------------------------------------------------------------
__global__ __launch_bounds__(256)
void lm_head_wmma_kernel(const float* __restrict__ X,
                         const float* __restrict__ Wlm,
                         const float* __restrict__ blm,
                         float* __restrict__ logits) {
    const int wave = threadIdx.x >> 5;
    const int lane = threadIdx.x & 31;
    const int m0   = blockIdx.x * 16;                 // row-tile base (0..112)
    const int n_tile = blockIdx.y * 8 + wave;
    const int n0   = n_tile * 16;
    if (n0 >= VOCAB) return;                          // wave-uniform exit

    const int hi = lane >> 4;                         // lane half: K pair select
    const int lm = lane & 15;

    // A: 16x4 f32 layout — lanes 0-15 hold K={0,1}, lanes 16-31 hold K={2,3}
    v2f a[8];
    #pragma unroll
    for (int s = 0; s < 8; ++s) {
        const float* xr = X + (m0 + lm) * EMBED + 4 * s + 2 * hi;
        a[s].x = xr[0];
        a[s].y = xr[1];
    }

    const int  col  = n0 + lm;
    const bool ok   = col < VOCAB;
    const int  colc = ok ? col : (VOCAB - 1);         // clamp: keep EXEC convergent

    v8f c = {};
    #pragma unroll
    for (int s = 0; s < 8; ++s) {
        // B: 4x16 f32 — mirror of A: lanes 0-15 rows {4s,4s+1}, 16-31 rows {4s+2,4s+3}
        const int krow = 4 * s + 2 * hi;
        v2f bf;
        bf.x = Wlm[(size_t)krow       * VOCAB + colc];
        bf.y = Wlm[(size_t)(krow + 1) * VOCAB + colc];
        c = __builtin_amdgcn_wmma_f32_16x16x4_f32(
                /*neg_a=*/false, a[s], /*neg_b=*/false, bf,
                /*c_mod=*/(short)0, c, /*reuse_a=*/false, /*reuse_b=*/false);
    }

    const float bias = blm[colc];
    #pragma unroll
    for (int r = 0; r < 8; ++r) {                     // VGPR r -> rows r, r+8
        const int row = m0 + r + 8 * hi;
        if (ok) logits[(size_t)row * VOCAB + col] = c[r] + bias;
    }
}

// ---------------------------------------------------------------------------
// Kernel 3: loss = -mean(log_softmax(logits)[targets]); one block per row.
// ---------------------------------------------------------------------------
__global__ __launch_bounds__(256)
void loss_kernel(const float* __restrict__ logits,
                 const int* __restrict__ targets,
                 float* __restrict__ loss_out) {
    __shared__ float red[256];
    const int row = blockIdx.x;
    const int tid = threadIdx.x;
    const float* lr = logits + (size_t)row * VOCAB;

    float mx = -INFINITY;
    for (int j = tid; j < VOCAB; j += 256) mx = fmaxf(mx, lr[j]);
    red[tid] = mx; __syncthreads();
    for (int s = 128; s > 0; s >>= 1) {
        if (tid < s) red[tid] = fmaxf(red[tid], red[tid + s]);
        __syncthreads();
    }
    mx = red[0];
    __syncthreads();

    float sum = 0.f;
    for (int j = tid; j < VOCAB; j += 256) sum += __expf(lr[j] - mx);
    red[tid] = sum; __syncthreads();
    for (int s = 128; s > 0; s >>= 1) {
        if (tid < s) red[tid] += red[tid + s];
        __syncthreads();
    }
    if (tid == 0) {
        const float logZ = mx + __logf(red[0]);
        const float lp   = lr[targets[row]] - logZ;
        atomicAdd(loss_out, -lp / (float)ROWS);
    }
}

// ---------------------------------------------------------------------------
extern "C" void kernel_launch(void* const* d_in, const int* in_sizes, int n_in,
                              void* d_out, int out_size, void* d_ws, size_t ws_size,
                              hipStream_t stream) {
    const int*   idx     = (const int*)  d_in[0];
    const int*   targets = (const int*)  d_in[1];
    const float* tok_emb = (const float*)d_in[2];
    const float* pos_emb = (const float*)d_in[3];
    const float* Wq      = (const float*)d_in[4];
    const float* Wk      = (const float*)d_in[5];
    const float* Wv      = (const float*)d_in[6];
    const float* W_lm    = (const float*)d_in[7];
    const float* b_lm    = (const float*)d_in[8];

    float* logits = (float*)d_out;                       // [128, 50257]
    float* loss   = logits + (size_t)ROWS * VOCAB;       // scalar slot
    float* X      = (float*)d_ws;                        // [128, 32] attention out

    embed_attn_kernel<<<BATCH, 128, 0, stream>>>(idx, tok_emb, pos_emb,
                                                 Wq, Wk, Wv, X, loss);

    dim3 g2(ROWS / 16, (N_TILES + 7) / 8);               // (8, 393)
    lm_head_wmma_kernel<<<g2, 256, 0, stream>>>(X, W_lm, b_lm, logits);

    loss_kernel<<<ROWS, 256, 0, stream>>>(logits, targets, loss);
}